// NUFFT_22565758173802
// MI455X (gfx1250) — compile-verified
//
#include <hip/hip_runtime.h>
#include <math.h>

// ---------------------------------------------------------------------------
// NUFFT type-2:  kdata[c,p] = sum_{3x3} wy*wx * kgrid[c, iy, ix]
// kgrid = centered-DFT(apodized, padded image) done as  M * img * M  with
// v_wmma_f32_16x16x4_f32 complex GEMMs (M is the symmetric centered DFT
// matrix, padded 360 -> 368 = 23*16).
// ---------------------------------------------------------------------------

#define OSR   360        // oversampled grid (1.125 * 320)
#define OSZ   368        // padded to multiple of 16 (23 tiles)
#define NIMG  320
#define NCH   16
#define NPTS  262144
#define P0    20         // (360-320)/2

#define BETA_F   4.418130194f      // pi*sqrt((3/1.125)^2*(0.625)^2 - 0.8)
#define BETA2_F  19.519874f        // BETA^2
#define DFT_SCL  0.052704627f      // 1/sqrt(360)  (applied twice -> 1/360 ortho)

typedef __attribute__((ext_vector_type(2))) float v2f;
typedef __attribute__((ext_vector_type(8))) float v8f;

// Modified Bessel I0 (Abramowitz & Stegun 9.8.1 / 9.8.2), t >= 0
static __device__ __forceinline__ float i0f(float t) {
    if (t < 3.75f) {
        float y = t * (1.0f / 3.75f);
        y *= y;
        return 1.0f + y * (3.5156229f + y * (3.0899424f + y * (1.2067492f +
                     y * (0.2659732f + y * (0.0360768f + y * 0.0045813f)))));
    } else {
        float y = 3.75f / t;
        float p = 0.39894228f + y * (0.01328592f + y * (0.00225319f +
                  y * (-0.00157565f + y * (0.00916281f + y * (-0.02057706f +
                  y * (0.02635537f + y * (-0.01647633f + y * 0.00392377f)))))));
        return p * expf(t) * rsqrtf(t);
    }
}

// image-domain apodization correction: t/sinh(t), t = sqrt(B^2-(pi*W*idx/OS)^2)
static __device__ __forceinline__ float apodf(int n) {
    float idx = (float)(n - NIMG / 2);
    float u   = 0.026179938779914944f * idx;   // pi*3/360
    float t   = sqrtf(fmaxf(BETA2_F - u * u, 1e-12f));
    return t / sinhf(t);
}

// Kaiser-Bessel interpolation weight at distance d (|d| <= 1.5 by construction)
static __device__ __forceinline__ float kbwf(float d) {
    float x   = d * (2.0f / 3.0f);
    float arg = fmaxf(1.0f - x * x, 0.0f);
    return i0f(BETA_F * sqrtf(arg));
}

// ---------------------------------------------------------------------------
// Build centered DFT matrix M[k,n] = exp(-2*pi*i*(k-180)(n-180)/360)/sqrt(360)
// Padded region (>=360) is zero. Exact integer phase reduction mod 360.
// ---------------------------------------------------------------------------
__global__ void build_dft_kernel(float* __restrict__ Mr, float* __restrict__ Mi) {
    int n = blockIdx.x * blockDim.x + threadIdx.x;   // 0 .. 368*368-1
    int k = n / OSZ, j = n % OSZ;
    float cr = 0.0f, ci = 0.0f;
    if (k < OSR && j < OSR) {
        int e = ((k - 180) * (j - 180)) % 360;
        if (e < 0) e += 360;
        float ang = (float)e * 0.017453292519943295f;   // pi/180
        float s, c;
        sincosf(ang, &s, &c);
        cr =  c * DFT_SCL;
        ci = -s * DFT_SCL;
    }
    Mr[n] = cr;
    Mi[n] = ci;
}

// ---------------------------------------------------------------------------
// Apodize + zero-pad image into 368x368 split real/imag planes per channel.
// ---------------------------------------------------------------------------
__global__ void prep_image_kernel(const float* __restrict__ re,
                                  const float* __restrict__ im,
                                  float* __restrict__ Ir,
                                  float* __restrict__ Ii) {
    long long n = (long long)blockIdx.x * blockDim.x + threadIdx.x;
    const long long plane = (long long)OSZ * OSZ;
    int c = (int)(n / plane);
    int r = (int)(n % plane);
    int y = r / OSZ, x = r % OSZ;
    float vr = 0.0f, vi = 0.0f;
    int sy = y - P0, sx = x - P0;
    if (sy >= 0 && sy < NIMG && sx >= 0 && sx < NIMG) {
        float a = apodf(sy) * apodf(sx);
        long long src = ((long long)c * NIMG + sy) * NIMG + sx;
        vr = re[src] * a;
        vi = im[src] * a;
    }
    Ir[n] = vr;
    Ii[n] = vi;
}

// ---------------------------------------------------------------------------
// Complex GEMM  D = A * B  (368x368x368), one wave per 16x16 output tile.
// 4 real WMMA accumulator chains: Dr = ArBr - AiBi, Di = ArBi + AiBr.
// INTERLEAVED=false -> split-plane output (Dr/Di); true -> float2 interleaved
// complex written through Dr.
// ---------------------------------------------------------------------------
template <bool INTERLEAVED>
__global__ __launch_bounds__(32)
void dft_cgemm_kernel(const float* __restrict__ Ar, const float* __restrict__ Ai,
                      long long sA,
                      const float* __restrict__ Br, const float* __restrict__ Bi,
                      long long sB,
                      float* __restrict__ Dr, float* __restrict__ Di,
                      long long sD) {
    const int lane = threadIdx.x;       // 0..31, EXEC all ones
    const int l15  = lane & 15;
    const int kp   = lane >> 4;         // 0 or 1
    const int rm   = blockIdx.x * 16;
    const int cn   = blockIdx.y * 16;
    const int ch   = blockIdx.z;

    const float* ar = Ar + (long long)ch * sA;
    const float* ai = Ai + (long long)ch * sA;
    const float* br = Br + (long long)ch * sB;
    const float* bi = Bi + (long long)ch * sB;

    // A-matrix: lane holds row (rm+l15), k = kb + 2*kp + {0,1}
    const float* aRowR = ar + (long long)(rm + l15) * OSZ;
    const float* aRowI = ai + (long long)(rm + l15) * OSZ;

    v8f rr = {}, ii = {}, ri = {}, ir = {};

    for (int kb = 0; kb < OSZ; kb += 4) {
        int k0 = kb + kp * 2;
        v2f aR = *(const v2f*)(aRowR + k0);
        v2f aI = *(const v2f*)(aRowI + k0);
        v2f bR, bI;
        long long b0 = (long long)k0 * OSZ + cn + l15;
        bR.x = br[b0];
        bR.y = br[b0 + OSZ];
        bI.x = bi[b0];
        bI.y = bi[b0 + OSZ];
        rr = __builtin_amdgcn_wmma_f32_16x16x4_f32(false, aR, false, bR, (short)0, rr, false, false);
        ii = __builtin_amdgcn_wmma_f32_16x16x4_f32(false, aI, false, bI, (short)0, ii, false, false);
        ri = __builtin_amdgcn_wmma_f32_16x16x4_f32(false, aR, false, bI, (short)0, ri, false, false);
        ir = __builtin_amdgcn_wmma_f32_16x16x4_f32(false, aI, false, bR, (short)0, ir, false, false);
    }

    // C/D layout: VGPR r -> row r + 8*kp, col = lane&15
#pragma unroll
    for (int r = 0; r < 8; ++r) {
        int row = rm + r + 8 * kp;
        int col = cn + l15;
        float dre = rr[r] - ii[r];
        float dim_ = ri[r] + ir[r];
        if (INTERLEAVED) {
            long long o = ((long long)ch * sD + (long long)row * OSZ + col) * 2;
            Dr[o]     = dre;
            Dr[o + 1] = dim_;
        } else {
            long long o = (long long)ch * sD + (long long)row * OSZ + col;
            Dr[o] = dre;
            Di[o] = dim_;
        }
    }
}

// ---------------------------------------------------------------------------
// 3x3 Kaiser-Bessel gather. G is interleaved complex [NCH][368*368] float2.
// Output: complex64 (C,P) flattened -> interleaved floats.
// ---------------------------------------------------------------------------
__global__ void interp_kernel(const float* __restrict__ coord,
                              const float* __restrict__ G,
                              float* __restrict__ out) {
    int p = blockIdx.x * blockDim.x + threadIdx.x;
    if (p >= NPTS) return;

    float py = coord[2 * p + 0] * 1.125f + 180.0f;   // OS/N = 1.125, OS//2 = 180
    float px = coord[2 * p + 1] * 1.125f + 180.0f;
    float sy = ceilf(py - 1.5f);
    float sx = ceilf(px - 1.5f);

    float wy[3], wx[3];
    int   iy[3], ix[3];
#pragma unroll
    for (int t = 0; t < 3; ++t) {
        float gy = sy + (float)t;
        float gx = sx + (float)t;
        wy[t] = kbwf(py - gy);
        wx[t] = kbwf(px - gx);
        int my = ((int)gy) % OSR; if (my < 0) my += OSR;
        int mx = ((int)gx) % OSR; if (mx < 0) mx += OSR;
        iy[t] = my;
        ix[t] = mx;
    }

    int   off[9];
    float wgt[9];
#pragma unroll
    for (int a = 0; a < 3; ++a)
#pragma unroll
        for (int b = 0; b < 3; ++b) {
            off[a * 3 + b] = iy[a] * OSZ + ix[b];
            wgt[a * 3 + b] = wy[a] * wx[b];
        }

    const long long plane = (long long)OSZ * OSZ;
    const float2* g2 = (const float2*)G;
#pragma unroll 1
    for (int c = 0; c < NCH; ++c) {
        const float2* gc = g2 + (long long)c * plane;
        float ar = 0.0f, ai = 0.0f;
#pragma unroll
        for (int t = 0; t < 9; ++t) {
            float2 v = gc[off[t]];
            ar = fmaf(wgt[t], v.x, ar);
            ai = fmaf(wgt[t], v.y, ai);
        }
        long long o = ((long long)c * NPTS + p) * 2;
        out[o]     = ar;
        out[o + 1] = ai;
    }
}

// ---------------------------------------------------------------------------
extern "C" void kernel_launch(void* const* d_in, const int* in_sizes, int n_in,
                              void* d_out, int out_size, void* d_ws, size_t ws_size,
                              hipStream_t stream) {
    const float* img_r = (const float*)d_in[0];   // (16,320,320) f32
    const float* img_i = (const float*)d_in[1];   // (16,320,320) f32
    const float* coord = (const float*)d_in[2];   // (P,2) f32
    float* out = (float*)d_out;                   // (16,P) complex64 interleaved

    const long long plane = (long long)OSZ * OSZ;   // 135424
    // workspace layout (floats): 98 planes total ~= 53.1 MB
    float* Mr = (float*)d_ws;            // [plane]
    float* Mi = Mr + plane;              // [plane]
    float* Ir = Mi + plane;              // [16*plane]
    float* Ii = Ir + 16 * plane;         // [16*plane]
    float* Tr = Ii + 16 * plane;         // [16*plane]
    float* Ti = Tr + 16 * plane;         // [16*plane]
    float* Gc = Ti + 16 * plane;         // [16*plane*2] interleaved complex

    // 1. centered DFT matrix (368*368 = 529 * 256)
    build_dft_kernel<<<529, 256, 0, stream>>>(Mr, Mi);

    // 2. apodize + pad image
    prep_image_kernel<<<16 * 529, 256, 0, stream>>>(img_r, img_i, Ir, Ii);

    // 3. T = M * img   (split-plane output)
    dim3 gg(OSZ / 16, OSZ / 16, NCH);
    dft_cgemm_kernel<false><<<gg, 32, 0, stream>>>(Mr, Mi, 0LL,
                                                   Ir, Ii, plane,
                                                   Tr, Ti, plane);

    // 4. G = T * M     (interleaved complex output; M symmetric)
    dft_cgemm_kernel<true><<<gg, 32, 0, stream>>>(Tr, Ti, plane,
                                                  Mr, Mi, 0LL,
                                                  Gc, nullptr, plane);

    // 5. 3x3 KB gather -> kdata
    interp_kernel<<<NPTS / 256, 256, 0, stream>>>(coord, Gc, out);
}